// EMA_76699525972315
// MI455X (gfx1250) — compile-verified
//
#include <hip/hip_runtime.h>

// EMA scan y_t = d*x_t + (1-d)*y_{t-1}, y_{-1} = x[0], over n float32 (n = 2^24).
// Memory-bound (~128 MB @ 23.3 TB/s => ~5.6 us floor): parallelized via
// carry-truncation halo ((1-d)^256 == 0 in f32 for d=0.9).
// Intra-tile scan runs on the CDNA5 matrix path: V_WMMA_F32_16X16X4_F32.

typedef __attribute__((ext_vector_type(2))) float v2f;
typedef __attribute__((ext_vector_type(8))) float v8f;

#define TILE 256              // elements per WMMA tile (16x16 matrix)
#define TILES_PER_REGION 16   // 4096 elements owned per wave
#define HALO_TILES 1          // 256-element halo rebuilds carry: g^256 -> 0
#define REGION (TILE * TILES_PER_REGION)

__global__ __launch_bounds__(256)
void ema_wmma_scan(const float* __restrict__ x,
                   const float* __restrict__ decay_p,
                   float* __restrict__ y,
                   int nRegions) {
  const int wavesPerBlock = blockDim.x >> 5;
  const int region = blockIdx.x * wavesPerBlock + ((int)threadIdx.x >> 5);
  if (region >= nRegions) return;          // wave-uniform: EXEC stays all-ones
  const int lane = (int)threadIdx.x & 31;
  const int n16  = lane & 15;              // column N / row M index
  const int hi   = lane >> 4;              // lane-half select

  const float d  = decay_p[0];
  const float g  = 1.0f - d;
  const float g2 = g * g, g4 = g2 * g2, g8 = g4 * g4;
  const float G16 = g8 * g8;                                 // g^16 (per-column)
  const float G16_2 = G16 * G16, G16_4 = G16_2 * G16_2;
  const float G16_8 = G16_4 * G16_4, G16_16 = G16_8 * G16_8; // g^256 (per-tile)

  // A-matrix (scan operator L, L[m][j] = d * g^(m-j), lower triangular), 4 K-chunks.
  // A layout: lanes = M rows (both halves), VGPR0 = {K0|K2}, VGPR1 = {K1|K3}.
  const int m = n16;
  v2f A[4];
#pragma unroll
  for (int k = 0; k < 4; ++k) {
    const int j0 = 4 * k + 2 * hi;
    float a0 = 0.0f, a1 = 0.0f;
    if (j0 <= m)     { float r = d; for (int t = 0; t < m - j0;     ++t) r *= g; a0 = r; }
    if (j0 + 1 <= m) { float r = d; for (int t = 0; t < m - j0 - 1; ++t) r *= g; a1 = r; }
    A[k].x = a0; A[k].y = a1;
  }

  // Carry multipliers per output row: g^(M+1), M = v + 8*hi (C/D layout).
  float gcol[8];
  {
    float p = g;
    for (int t = 0; t < 8 * hi; ++t) p *= g;
#pragma unroll
    for (int v = 0; v < 8; ++v) { gcol[v] = p; p *= g; }
  }
  // G16^(column index) for injecting the running carry into each column.
  float powG16c = 1.0f;
  for (int t = 0; t < n16; ++t) powG16c *= G16;

  const long regionStart = (long)region * REGION;
  float carry;
  int t0;
  if (region == 0) { carry = x[0]; t0 = 0; }              // y_{-1} = x[0]
  else             { carry = 0.0f; t0 = -HALO_TILES; }    // halo rebuilds carry

  for (int t = t0; t < TILES_PER_REGION; ++t) {
    const float* tb = x + regionStart + (long)t * TILE;
    __builtin_prefetch(tb + 2 * TILE, 0, 1);              // global_prefetch_b8

    // B layout (4x16): lane = column N, low half K={4k,4k+1}, high half K={4k+2,4k+3}
    // -> one aligned b64 load per chunk, directly in-register-layout.
    v8f acc = {};
#pragma unroll
    for (int k = 0; k < 4; ++k) {
      const float* p = tb + 16 * n16 + 4 * k + 2 * hi;
      v2f b = *(const v2f*)p;
      acc = __builtin_amdgcn_wmma_f32_16x16x4_f32(
          false, A[k], false, b, (short)0, acc, false, false);
    }

    // Zero-carry last row of each column lives in acc[7] of lanes 16..31.
    float ylast0 = __shfl(acc[7], 16 + n16);
    // Weighted inclusive scan over 16 columns: S_c = ylast0_c + G16 * S_{c-1}.
    float S = ylast0;
    float f = G16;
#pragma unroll
    for (int off = 1; off < 16; off <<= 1) {
      float tt = __shfl_up(S, off, 16);
      S += (n16 >= off) ? f * tt : 0.0f;
      f *= f;
    }
    float wc = __shfl_up(S, 1, 16);
    float wcarry = ((n16 == 0) ? 0.0f : wc) + powG16c * carry;

    if (t >= 0) {                                         // halo tile: carry only
      // Build the fixed-up tile as a fresh 32-byte value so the backend emits
      // exactly two aligned global_store_b128 (lane address is 32B-aligned).
      v8f outv;
#pragma unroll
      for (int v = 0; v < 8; ++v)
        outv[v] = __builtin_fmaf(gcol[v], wcarry, acc[v]);
      float* yb = y + regionStart + (long)t * TILE + 16 * n16 + 8 * hi;
      *reinterpret_cast<v8f*>(yb) = outv;
    }
    carry = __shfl(S, 15) + G16_16 * carry;               // carry across tiles
  }
}

extern "C" void kernel_launch(void* const* d_in, const int* in_sizes, int n_in,
                              void* d_out, int out_size, void* d_ws, size_t ws_size,
                              hipStream_t stream) {
  const float* x     = (const float*)d_in[0];
  const float* decay = (const float*)d_in[1];
  float* out         = (float*)d_out;
  const int n = in_sizes[0];                // 2^24, divisible by REGION
  int nRegions = n / REGION;
  if (nRegions < 1) nRegions = 1;
  const int wavesPerBlock = 8;              // 256 threads = 8 wave32
  const int blocks = (nRegions + wavesPerBlock - 1) / wavesPerBlock;
  ema_wmma_scan<<<blocks, 256, 0, stream>>>(x, decay, out, nRegions);
}